// GNNEncoder_19559281066476
// MI455X (gfx1250) — compile-verified
//
#include <hip/hip_runtime.h>
#include <hip/hip_bf16.h>

typedef float v2f __attribute__((ext_vector_type(2)));
typedef float v8f __attribute__((ext_vector_type(8)));

#define NEG_SLOPE 0.2f
#define NEG_INF_KEY 0x007FFFFFu  // f2key(-inf)

__device__ __forceinline__ unsigned f2key(float f) {
  unsigned u = __float_as_uint(f);
  return (u & 0x80000000u) ? ~u : (u | 0x80000000u);
}
__device__ __forceinline__ float key2f(unsigned k) {
  unsigned u = (k & 0x80000000u) ? (k ^ 0x80000000u) : ~k;
  return __uint_as_float(u);
}

// ---- fill helper -----------------------------------------------------------
__global__ void fill_u32(unsigned* __restrict__ p, unsigned v, int n) {
  int i = blockIdx.x * blockDim.x + threadIdx.x;
  if (i < n) p[i] = v;
}

// ---- v = We @ att_edge (384-vector), both layers ---------------------------
__global__ void build_vedge(const float* __restrict__ We1, const float* __restrict__ ae1,
                            const float* __restrict__ We2, const float* __restrict__ ae2,
                            float* __restrict__ v1, float* __restrict__ v2,
                            int SB, int C1, int C2) {
  int i = blockIdx.x * blockDim.x + threadIdx.x;
  if (i >= SB) return;
  float s = 0.f;
  const float* r1 = We1 + (size_t)i * C1;
  for (int c = 0; c < C1; ++c) s += r1[c] * ae1[c];
  v1[i] = s;
  s = 0.f;
  const float* r2 = We2 + (size_t)i * C2;
  for (int c = 0; c < C2; ++c) s += r2[c] * ae2[c];
  v2[i] = s;
}

// ---- single fused pass over edge_attr: a_e1[e], a_e2[e] (wave per edge) ----
__global__ void edge_dots(const float* __restrict__ ea,
                          const float* __restrict__ v1, const float* __restrict__ v2,
                          float* __restrict__ ae1, float* __restrict__ ae2,
                          int E, int SB) {
  int wave = threadIdx.x >> 5, lane = threadIdx.x & 31;
  int e = blockIdx.x * (blockDim.x >> 5) + wave;
  if (e >= E) return;
  const float* row = ea + (size_t)e * SB;
  float s1 = 0.f, s2 = 0.f;
  for (int j = lane; j < SB; j += 32) {
    float t = row[j];
    s1 += t * v1[j];
    s2 += t * v2[j];
  }
  for (int off = 16; off > 0; off >>= 1) {
    s1 += __shfl_down(s1, off, 32);
    s2 += __shfl_down(s2, off, 32);
  }
  if (lane == 0) { ae1[e] = s1; ae2[e] = s2; }
}

// ---- f32 WMMA GEMM: C[M,Nc] = A[M,K] @ B[K,Nc]; optional row gather --------
// One wave -> one 16x16 C tile, K stepped by 4 via V_WMMA_F32_16X16X4_F32.
__global__ void gemm16_wmma(const float* __restrict__ A, const int* __restrict__ xidx,
                            const float* __restrict__ B, float* __restrict__ C,
                            int M, int K, int Nc) {
  const int wave = threadIdx.x >> 5;
  const int lane = threadIdx.x & 31;
  const int m0 = (blockIdx.x * (blockDim.x >> 5) + wave) << 4;
  const int n0 = blockIdx.y << 4;
  if (m0 >= M) return;                 // wave-uniform: EXEC stays all-ones
  const int hi = lane >> 4;            // lane half selects K pair k0+2*hi, k0+2*hi+1
  const int l15 = lane & 15;
  int row = m0 + l15;
  if (row >= M) row = M - 1;           // clamp; tail masked at store
  const float* arow = xidx ? (A + (size_t)xidx[row] * K) : (A + (size_t)row * K);
  const float* bcol = B + n0 + l15;
  v8f acc = {};
  for (int k0 = 0; k0 < K; k0 += 4) {
    const int ka = k0 + 2 * hi;
    v2f a, b;
    a.x = arow[ka];                       // A[m][ka]   -> VGPR0
    a.y = arow[ka + 1];                   // A[m][ka+1] -> VGPR1
    b.x = bcol[(size_t)ka * Nc];          // B[ka][n]
    b.y = bcol[(size_t)(ka + 1) * Nc];    // B[ka+1][n]
    acc = __builtin_amdgcn_wmma_f32_16x16x4_f32(false, a, false, b,
                                                (short)0, acc, false, false);
  }
  float* cp = C + (size_t)(m0 + 8 * hi) * Nc + n0 + l15;
#pragma unroll
  for (int r = 0; r < 8; ++r)
    if (m0 + 8 * hi + r < M) cp[(size_t)r * Nc] = acc[r];
}

// ---- per-node attention scalars: a_src[n], a_dst[n] ------------------------
__global__ void node_scalars(const float* __restrict__ hs,
                             const float* __restrict__ att_s, const float* __restrict__ att_d,
                             float* __restrict__ as, float* __restrict__ ad, int N, int C) {
  int n = blockIdx.x * blockDim.x + threadIdx.x;
  if (n >= N) return;
  const float* r = hs + (size_t)n * C;
  float s = 0.f, d = 0.f;
  for (int c = 0; c < C; ++c) { float t = r[c]; s += t * att_s[c]; d += t * att_d[c]; }
  as[n] = s; ad[n] = d;
}

// ---- edge pass 1: alpha = leaky_relu(...), segment max (ordered-key) -------
__global__ void edge_alpha_max(const int* __restrict__ src, const int* __restrict__ dst,
                               const float* __restrict__ ae, const float* __restrict__ as,
                               const float* __restrict__ ad, float* __restrict__ alpha,
                               unsigned* __restrict__ mkey, int E) {
  int e = blockIdx.x * blockDim.x + threadIdx.x;
  if (e >= E) return;
  int s = src[e], d = dst[e];
  float a = as[s] + ad[d] + ae[e];
  a = (a >= 0.f) ? a : NEG_SLOPE * a;
  alpha[e] = a;
  atomicMax(&mkey[d], f2key(a));
}

// ---- edge pass 2: e = exp(alpha - m[dst]); segment sum ---------------------
__global__ void edge_exp_sum(const int* __restrict__ dst, float* __restrict__ alpha,
                             const unsigned* __restrict__ mkey, float* __restrict__ ssum,
                             int E) {
  int e = blockIdx.x * blockDim.x + threadIdx.x;
  if (e >= E) return;
  int d = dst[e];
  float ex = expf(alpha[e] - key2f(mkey[d]));
  alpha[e] = ex;                 // in-place: alpha buffer now holds exp values
  atomicAdd(&ssum[d], ex);
}

// ---- edge pass 3: scatter messages hs[src]*coef into agg[dst] (wave/edge) --
__global__ void edge_scatter(const int* __restrict__ src, const int* __restrict__ dst,
                             const float* __restrict__ eexp, const float* __restrict__ ssum,
                             const float* __restrict__ hs, float* __restrict__ agg,
                             int E, int C) {
  int wave = threadIdx.x >> 5, lane = threadIdx.x & 31;
  int e = blockIdx.x * (blockDim.x >> 5) + wave;
  if (e >= E) return;
  int s = src[e], d = dst[e];
  float coef = eexp[e] / (ssum[d] + 1e-16f);
  const float* hr = hs + (size_t)s * C;
  float* ar = agg + (size_t)d * C;
  for (int c = lane; c < C; c += 32) atomicAdd(&ar[c], hr[c] * coef);
}

// ---- epilogues -------------------------------------------------------------
__global__ void epilogue_relu_bias(float* __restrict__ h, const float* __restrict__ b,
                                   int n, int C) {
  int i = blockIdx.x * blockDim.x + threadIdx.x;
  if (i >= n) return;
  float v = h[i] + b[i % C];
  h[i] = v > 0.f ? v : 0.f;
}
__global__ void epilogue_bias(float* __restrict__ h, const float* __restrict__ b,
                              int n, int C) {
  int i = blockIdx.x * blockDim.x + threadIdx.x;
  if (i >= n) return;
  h[i] = h[i] + b[i % C];
}

extern "C" void kernel_launch(void* const* d_in, const int* in_sizes, int n_in,
                              void* d_out, int out_size, void* d_ws, size_t ws_size,
                              hipStream_t stream) {
  const int N  = in_sizes[0];
  const int E  = in_sizes[1] / 2;
  const int SB = in_sizes[2] / E;      // 384
  const int C1 = in_sizes[6];          // 32
  const int C2 = in_sizes[12];         // 128
  const int K0 = in_sizes[4] / C1;     // 128 (author emb dim)

  const int*   x    = (const int*)d_in[0];
  const int*   src  = (const int*)d_in[1];
  const int*   dst  = src + E;
  const float* ea   = (const float*)d_in[2];
  const float* emb  = (const float*)d_in[3];
  const float* W1   = (const float*)d_in[4];
  const float* We1  = (const float*)d_in[5];
  const float* as1  = (const float*)d_in[6];
  const float* ad1  = (const float*)d_in[7];
  const float* aev1 = (const float*)d_in[8];
  const float* b1   = (const float*)d_in[9];
  const float* W2   = (const float*)d_in[10];
  const float* We2  = (const float*)d_in[11];
  const float* as2  = (const float*)d_in[12];
  const float* ad2  = (const float*)d_in[13];
  const float* aev2 = (const float*)d_in[14];
  const float* b2   = (const float*)d_in[15];
  float* out = (float*)d_out;

  // workspace layout (floats)
  float* w     = (float*)d_ws;
  float* v1    = w;  w += SB;
  float* v2    = w;  w += SB;
  float* aE1   = w;  w += E;
  float* aE2   = w;  w += E;
  float* alpha = w;  w += E;                     // reused as exp(alpha)
  float* hs1   = w;  w += (size_t)N * C1;
  float* h1    = w;  w += (size_t)N * C1;        // agg1, then relu'd in place
  float* hs2   = w;  w += (size_t)N * C2;
  float* aS    = w;  w += N;
  float* aD    = w;  w += N;
  float* ssum  = w;  w += N;
  unsigned* mkey = (unsigned*)w;  w += N;

  const int B256 = 256;
  const int WPB  = 4;   // waves per 128-thread block
  dim3 blk128(128);

  // 1) v1/v2 = We @ att_edge  (kills the E x 384 x C edge GEMMs entirely)
  build_vedge<<<(SB + 127) / 128, 128, 0, stream>>>(We1, aev1, We2, aev2, v1, v2, SB, C1, C2);

  // 2) single 768MB pass over edge_attr -> both layers' edge scalars
  edge_dots<<<(E + WPB - 1) / WPB, blk128, 0, stream>>>(ea, v1, v2, aE1, aE2, E, SB);

  // 3) hs1 = emb[x] @ W1  (WMMA f32, gathered A rows)
  {
    int mt = (N + 15) / 16;
    dim3 g((mt + WPB - 1) / WPB, C1 / 16);
    gemm16_wmma<<<g, blk128, 0, stream>>>(emb, x, W1, hs1, N, K0, C1);
  }

  // 4) per-node scalars, layer 1
  node_scalars<<<(N + B256 - 1) / B256, B256, 0, stream>>>(hs1, as1, ad1, aS, aD, N, C1);

  // 5) init segment-max keys, segment sums, aggregation buffer
  fill_u32<<<(N + B256 - 1) / B256, B256, 0, stream>>>(mkey, NEG_INF_KEY, N);
  fill_u32<<<(N + B256 - 1) / B256, B256, 0, stream>>>((unsigned*)ssum, 0u, N);
  fill_u32<<<((N * C1) + B256 - 1) / B256, B256, 0, stream>>>((unsigned*)h1, 0u, N * C1);

  // 6-8) segment softmax + message aggregation, layer 1
  edge_alpha_max<<<(E + B256 - 1) / B256, B256, 0, stream>>>(src, dst, aE1, aS, aD, alpha, mkey, E);
  edge_exp_sum<<<(E + B256 - 1) / B256, B256, 0, stream>>>(dst, alpha, mkey, ssum, E);
  edge_scatter<<<(E + WPB - 1) / WPB, blk128, 0, stream>>>(src, dst, alpha, ssum, hs1, h1, E, C1);

  // 9) h1 = relu(agg1 + b1)
  epilogue_relu_bias<<<((N * C1) + B256 - 1) / B256, B256, 0, stream>>>(h1, b1, N * C1, C1);

  // 10) hs2 = h1 @ W2 (WMMA f32)
  {
    int mt = (N + 15) / 16;
    dim3 g((mt + WPB - 1) / WPB, C2 / 16);
    gemm16_wmma<<<g, blk128, 0, stream>>>(h1, nullptr, W2, hs2, N, C1, C2);
  }

  // 11) per-node scalars, layer 2
  node_scalars<<<(N + B256 - 1) / B256, B256, 0, stream>>>(hs2, as2, ad2, aS, aD, N, C2);

  // 12) re-init max/sum, zero output accumulator
  fill_u32<<<(N + B256 - 1) / B256, B256, 0, stream>>>(mkey, NEG_INF_KEY, N);
  fill_u32<<<(N + B256 - 1) / B256, B256, 0, stream>>>((unsigned*)ssum, 0u, N);
  fill_u32<<<((N * C2) + B256 - 1) / B256, B256, 0, stream>>>((unsigned*)out, 0u, N * C2);

  // 13-15) segment softmax + aggregation, layer 2 (accumulate into d_out)
  edge_alpha_max<<<(E + B256 - 1) / B256, B256, 0, stream>>>(src, dst, aE2, aS, aD, alpha, mkey, E);
  edge_exp_sum<<<(E + B256 - 1) / B256, B256, 0, stream>>>(dst, alpha, mkey, ssum, E);
  edge_scatter<<<(E + WPB - 1) / WPB, blk128, 0, stream>>>(src, dst, alpha, ssum, hs2, out, E, C2);

  // 16) out += b2
  epilogue_bias<<<((N * C2) + B256 - 1) / B256, B256, 0, stream>>>(out, b2, N * C2, C2);
}